// HGNNConv_78048145703099
// MI455X (gfx1250) — compile-verified
//
#include <hip/hip_runtime.h>
#include <hip/hip_bf16.h>

namespace {
constexpr int kN    = 100000;
constexpr int kE    = 25000;
constexpr int kNNZ  = 1600000;
constexpr int kCIN  = 256;
constexpr int kCOUT = 256;
}

typedef __attribute__((ext_vector_type(2))) float v2f;
typedef __attribute__((ext_vector_type(8))) float v8f;

// ---------------- degrees: dv[v]++ , de[e]++ (as float) ----------------
__global__ __launch_bounds__(256) void degree_kernel(const int* __restrict__ v_idx,
                                                     const int* __restrict__ e_idx,
                                                     float* __restrict__ dv,
                                                     float* __restrict__ de) {
  int i = blockIdx.x * blockDim.x + threadIdx.x;
  if (i < kNNZ) {
    atomicAdd(&dv[v_idx[i]], 1.0f);
    atomicAdd(&de[e_idx[i]], 1.0f);
  }
}

// in-place: d = d>0 ? rsqrt(d) : 0
__global__ __launch_bounds__(256) void rsqrt_kernel(float* __restrict__ d, int n) {
  int i = blockIdx.x * blockDim.x + threadIdx.x;
  if (i < n) {
    float x = d[i];
    d[i] = (x > 0.0f) ? rsqrtf(x) : 0.0f;
  }
}

// in-place: d = d>0 ? 1/d : 0
__global__ __launch_bounds__(256) void recip_kernel(float* __restrict__ d, int n) {
  int i = blockIdx.x * blockDim.x + threadIdx.x;
  if (i < n) {
    float x = d[i];
    d[i] = (x > 0.0f) ? (1.0f / x) : 0.0f;
  }
}

// ---------------- Xp = X @ W + b via V_WMMA_F32_16X16X4_F32 ----------------
// One 16x16 output tile per wave32. 4 waves / 128-thread block.
// A fragment (16x4 f32): lanes 0-15 -> M=lane, K={k,k+1}; lanes 16-31 -> M=lane-16, K={k+2,k+3}
// B fragment (4x16 f32): lanes 0-15 -> K={k,k+1} row, N=lane; lanes 16-31 -> K={k+2,k+3}, N=lane-16
// C/D (16x16 f32): VGPR j -> M=j (lanes 0-15) / M=j+8 (lanes 16-31), N=lane&15
__global__ __launch_bounds__(128) void gemm_wmma_f32(const float* __restrict__ X,
                                                     const float* __restrict__ W,
                                                     const float* __restrict__ b,
                                                     float* __restrict__ Xp) {
  const int wave = threadIdx.x >> 5;
  const int lane = threadIdx.x & 31;
  const int half = lane >> 4;   // 0: lanes 0-15, 1: lanes 16-31
  const int l16  = lane & 15;

  const int tile    = blockIdx.x * 4 + wave;   // total tiles = (N/16)*(COUT/16) = 100000
  const int rowTile = tile >> 4;               // COUT/16 == 16
  const int colTile = tile & 15;
  const int row0 = rowTile * 16;
  const int col0 = colTile * 16;

  const float* __restrict__ xrow = X + (size_t)(row0 + l16) * kCIN + 2 * half;
  const float* __restrict__ wcol = W + (size_t)(2 * half) * kCOUT + col0 + l16;

  v8f acc = {0.f, 0.f, 0.f, 0.f, 0.f, 0.f, 0.f, 0.f};

#pragma unroll 8
  for (int k = 0; k < kCIN; k += 4) {
    v2f a, bv;
    a.x  = xrow[k];
    a.y  = xrow[k + 1];
    bv.x = wcol[(size_t)k * kCOUT];
    bv.y = wcol[(size_t)(k + 1) * kCOUT];
    // 8 args: (neg_a, A, neg_b, B, c_mod, C, reuse_a, reuse_b)
    acc = __builtin_amdgcn_wmma_f32_16x16x4_f32(false, a, false, bv,
                                                (short)0, acc, false, false);
  }

  const float bias = b[col0 + l16];
#pragma unroll
  for (int j = 0; j < 8; ++j) {
    const int m = j + half * 8;
    Xp[(size_t)(row0 + m) * kCOUT + col0 + l16] = acc[j] + bias;
  }
}

// ---------------- Ye[e] += Xp[v] * dv_isqrt[v]  (atomic scatter) ----------------
// 64 lanes per pair, float4 per lane -> one coalesced 1KB row read per pair.
__global__ __launch_bounds__(256) void scatter_v2e(const float* __restrict__ Xp,
                                                   const float* __restrict__ dvis,
                                                   const int* __restrict__ v_idx,
                                                   const int* __restrict__ e_idx,
                                                   float* __restrict__ Ye) {
  const int pair = blockIdx.x * 4 + (threadIdx.x >> 6);
  const int c    = (threadIdx.x & 63) * 4;
  const int v = v_idx[pair];
  const int e = e_idx[pair];
  const float s = dvis[v];
  const float4 x = *(const float4*)(Xp + (size_t)v * kCOUT + c);
  float* y = Ye + (size_t)e * kCOUT + c;
  atomicAdd(y + 0, x.x * s);
  atomicAdd(y + 1, x.y * s);
  atomicAdd(y + 2, x.z * s);
  atomicAdd(y + 3, x.w * s);
}

// ---------------- Zv[v] += Ye[e] * de_inv[e]  (atomic scatter into d_out) ----------------
__global__ __launch_bounds__(256) void scatter_e2v(const float* __restrict__ Ye,
                                                   const float* __restrict__ deinv,
                                                   const int* __restrict__ v_idx,
                                                   const int* __restrict__ e_idx,
                                                   float* __restrict__ Zv) {
  const int pair = blockIdx.x * 4 + (threadIdx.x >> 6);
  const int c    = (threadIdx.x & 63) * 4;
  const int v = v_idx[pair];
  const int e = e_idx[pair];
  const float s = deinv[e];
  const float4 y = *(const float4*)(Ye + (size_t)e * kCOUT + c);
  float* z = Zv + (size_t)v * kCOUT + c;
  atomicAdd(z + 0, y.x * s);
  atomicAdd(z + 1, y.y * s);
  atomicAdd(z + 2, y.z * s);
  atomicAdd(z + 3, y.w * s);
}

// ---------------- out = relu(out * dv_isqrt[row]) ----------------
__global__ __launch_bounds__(256) void epilogue_relu(float* __restrict__ out,
                                                     const float* __restrict__ dvis) {
  const unsigned i = blockIdx.x * blockDim.x + threadIdx.x;
  if (i < (unsigned)kN * kCOUT) {
    const float val = out[i] * dvis[i / kCOUT];
    out[i] = val > 0.0f ? val : 0.0f;
  }
}

extern "C" void kernel_launch(void* const* d_in, const int* in_sizes, int n_in,
                              void* d_out, int out_size, void* d_ws, size_t ws_size,
                              hipStream_t stream) {
  (void)in_sizes; (void)n_in; (void)ws_size;

  const float* X   = (const float*)d_in[0];
  const float* W   = (const float*)d_in[1];
  const float* b   = (const float*)d_in[2];
  const int* v_idx = (const int*)d_in[3];
  const int* e_idx = (const int*)d_in[4];
  float* out = (float*)d_out;

  // workspace layout
  char* ws = (char*)d_ws;
  float* Xp = (float*)ws;                                           // N*COUT  = 102.4 MB
  float* Ye = (float*)(ws + (size_t)kN * kCOUT * sizeof(float));    // E*COUT  =  25.6 MB
  float* dv = (float*)(ws + ((size_t)kN * kCOUT + (size_t)kE * kCOUT) * sizeof(float)); // N floats
  float* de = dv + kN;                                              // E floats

  // zero accumulators (Ye, dv, de contiguous) and the Zv accumulator (d_out)
  const size_t zero_bytes = ((size_t)kE * kCOUT + kN + kE) * sizeof(float);
  hipMemsetAsync(Ye, 0, zero_bytes, stream);
  hipMemsetAsync(out, 0, (size_t)out_size * sizeof(float), stream);

  // degrees -> normalization factors (in place)
  degree_kernel<<<(kNNZ + 255) / 256, 256, 0, stream>>>(v_idx, e_idx, dv, de);
  rsqrt_kernel<<<(kN + 255) / 256, 256, 0, stream>>>(dv, kN);
  recip_kernel<<<(kE + 255) / 256, 256, 0, stream>>>(de, kE);

  // Xp = X @ W + b   (fp32 WMMA, 16x16 tile per wave, 4 waves/block)
  const int tiles = (kN / 16) * (kCOUT / 16);   // 100000
  gemm_wmma_f32<<<tiles / 4, 128, 0, stream>>>(X, W, b, Xp);

  // Ye = segsum_e(Xp[v] * dv^-1/2);  Zv = segsum_v(Ye[e] * de^-1)
  scatter_v2e<<<kNNZ / 4, 256, 0, stream>>>(Xp, dv, v_idx, e_idx, Ye);
  scatter_e2v<<<kNNZ / 4, 256, 0, stream>>>(Ye, de, v_idx, e_idx, out);

  // out = relu(Zv * dv^-1/2)
  epilogue_relu<<<(kN * kCOUT + 255) / 256, 256, 0, stream>>>(out, dv);
}